// ESN_81913616269486
// MI455X (gfx1250) — compile-verified
//
#include <hip/hip_runtime.h>
#include <hip/hip_bf16.h>

// ESN recurrence for MI455X (gfx1250, wave32).
//
// h_t = tanh(emb[:,t,:] @ Win^T + h_{t-1} @ W^T)          (pre-activation GEMM)
// h_t = 0.5*h_t + 0.5*h_{t-1}   for t >= 2                (leaky blend quirk)
//
// One kernel per timestep (2048 sequential launches, graph-captured).
// Concatenated-K GEMM per step: A = [x_t | h_{t-1}]  (32 x 1536)
//                               Bm = [Win ; W] rows   (1024 x 1536, row-major)
// Each WG: 32 output rows (all of B) x 32 output cols; 4 waves, one
// 16x16 f32 WMMA tile each. K is streamed through DOUBLE-BUFFERED LDS in
// chunks of 128 using GLOBAL_LOAD_ASYNC_TO_LDS_B128 (CDNA5 async copy,
// ASYNCcnt-tracked) so global latency hides behind the WMMA stream.

typedef float v2f __attribute__((ext_vector_type(2)));
typedef float v8f __attribute__((ext_vector_type(8)));

#define B_       32
#define T_       2048
#define D_       512
#define R_       1024
#define KC       128          // K chunk through LDS
#define LDT      132          // padded LDS row stride in floats (16B aligned, bank-spread)
#define NTHREADS 128
#define N_PER_WG 32
#define LEAKY    0.5f

// ---- CDNA5 async copy: 16B global -> LDS, no VGPR round-trip ------------
__device__ __forceinline__ void async_b128(const void* gsrc, void* ldst) {
    // VDST = per-lane LDS byte address (low 32 bits of generic shared ptr),
    // VADDR = 64-bit global address, GV mode (saddr = off).
    asm volatile("global_load_async_to_lds_b128 %0, %1, off"
                 :: "v"((unsigned int)(unsigned long long)ldst), "v"(gsrc)
                 : "memory");
}

__device__ __forceinline__ void wait_async0() {
#if __has_builtin(__builtin_amdgcn_s_wait_asynccnt)
    __builtin_amdgcn_s_wait_asynccnt(0);
#else
    asm volatile("s_wait_asynccnt 0" ::: "memory");
#endif
}

__device__ __forceinline__ float esn_tanh(float x) {
#if __has_builtin(__builtin_amdgcn_tanhf)
    return __builtin_amdgcn_tanhf(x);
#else
    return tanhf(x);
#endif
}

__global__ __launch_bounds__(NTHREADS)
void esn_step_kernel(const float* __restrict__ emb,   // [B, T, D]
                     const float* __restrict__ Win,   // [R, D]
                     const float* __restrict__ W,     // [R, R]
                     const float* __restrict__ h_prev,// [B, R] (== h_out for t==0, unused)
                     float* __restrict__ h_out,       // [B, R] slice of out[t]
                     int t)
{
    __shared__ float lA[2][B_ * LDT];        // A chunks: 32 x KC, double-buffered
    __shared__ float lB[2][N_PER_WG * LDT];  // B chunks: 32 x KC, double-buffered

    const int tid  = threadIdx.x;
    const int lane = tid & 31;
    const int wave = tid >> 5;
    const int n0   = blockIdx.x * N_PER_WG;

    // wave -> (M tile, N tile): M in {0,16}, N in {0,16} within WG
    const int m_off = (wave & 1) * 16;
    const int nt    = (wave >> 1) * 16;

    // f32 16x16x4 A/B fragment coords (ISA 7.12.2):
    //   lane 0-15 : row = lane,    K pair {0,1}
    //   lane 16-31: row = lane-16, K pair {2,3}
    const int fr = lane & 15;
    const int fk = (lane >> 4) * 2;

    // t == 0: h_{t-1} is all-zero -> skip the W·h half of K entirely.
    const int kmax    = (t == 0) ? D_ : (D_ + R_);
    const int nchunks = kmax / KC;

    // ---- async-issue one K chunk (A tile + B tile) into LDS buffer `buf`
    auto issue_chunk = [&](int k0, int buf) {
        const bool inEmb = (k0 < D_);        // uniform: 512 % 128 == 0
        float* la = &lA[buf][0];
        float* lb = &lB[buf][0];
        // A(b, k) = k < D ? emb[b][t][k] : h_prev[b][k-D]
        for (int i = tid; i < B_ * (KC / 4); i += NTHREADS) {
            const int row = i / (KC / 4);
            const int c4  = (i % (KC / 4)) * 4;
            const float* src = inEmb
                ? (emb + ((size_t)row * T_ + t) * D_ + (k0 + c4))
                : (h_prev + (size_t)row * R_ + (k0 - D_ + c4));
            async_b128(src, &la[row * LDT + c4]);
        }
        // Bm(n, k) = k < D ? Win[n0+n][k] : W[n0+n][k-D]   (row-major rows)
        for (int i = tid; i < N_PER_WG * (KC / 4); i += NTHREADS) {
            const int row = i / (KC / 4);
            const int c4  = (i % (KC / 4)) * 4;
            const float* src = inEmb
                ? (Win + (size_t)(n0 + row) * D_ + (k0 + c4))
                : (W   + (size_t)(n0 + row) * R_ + (k0 - D_ + c4));
            async_b128(src, &lb[row * LDT + c4]);
        }
    };

    v8f acc = {};   // 16x16 f32 accumulator (8 VGPRs)

    issue_chunk(0, 0);                       // prologue: fill buffer 0

    for (int c = 0; c < nchunks; ++c) {
        const int cur = c & 1;

        wait_async0();                       // my async writes (chunk c) landed
        __syncthreads();                     // everyone's writes visible

        if (c + 1 < nchunks)                 // prefetch next chunk into idle buf
            issue_chunk((c + 1) * KC, cur ^ 1);

        // ---- 32 x V_WMMA_F32_16X16X4_F32 over this K chunk ----
        const float* la = &lA[cur][0];
        const float* lb = &lB[cur][0];
        #pragma unroll
        for (int kk = 0; kk < KC; kk += 4) {
            v2f a = *(const v2f*)&la[(m_off + fr) * LDT + kk + fk];
            v2f b = *(const v2f*)&lb[(nt    + fr) * LDT + kk + fk];
            acc = __builtin_amdgcn_wmma_f32_16x16x4_f32(
                /*neg_a=*/false, a, /*neg_b=*/false, b,
                /*c_mod=*/(short)0, acc, /*reuse_a=*/false, /*reuse_b=*/false);
        }
        __syncthreads();                     // all reads of buf `cur` done
    }

    // ---- epilogue: tanh + leaky blend + store ----
    // C/D layout: VGPR p -> M = p + 8*(lane>=16); N = lane % 16
    const int n     = n0 + nt + (lane & 15);
    const int bbase = m_off + ((lane >> 4) * 8);
    const bool blend = (t >= 2);             // uniform
    #pragma unroll
    for (int p = 0; p < 8; ++p) {
        const int b = bbase + p;
        float h_new = esn_tanh(acc[p]);
        float out_v;
        if (blend) {
            const float hp = h_prev[(size_t)b * R_ + n];
            out_v = (1.0f - LEAKY) * h_new + LEAKY * hp;
        } else {
            out_v = h_new;
        }
        h_out[(size_t)b * R_ + n] = out_v;
    }
}

extern "C" void kernel_launch(void* const* d_in, const int* in_sizes, int n_in,
                              void* d_out, int out_size, void* d_ws, size_t ws_size,
                              hipStream_t stream) {
    const float* emb = (const float*)d_in[0];   // [B, T, D]
    const float* Win = (const float*)d_in[1];   // [R, D]
    const float* W   = (const float*)d_in[2];   // [R, R]
    float* out = (float*)d_out;                 // [T, B, R]

    for (int t = 0; t < T_; ++t) {
        const float* h_prev = (t == 0) ? out : (out + (size_t)(t - 1) * B_ * R_);
        float* h_out = out + (size_t)t * B_ * R_;
        esn_step_kernel<<<dim3(R_ / N_PER_WG), dim3(NTHREADS), 0, stream>>>(
            emb, Win, W, h_prev, h_out, t);
    }
}